// GraphConvModel_77111842833028
// MI455X (gfx1250) — compile-verified
//
#include <hip/hip_runtime.h>

#define FT 128
#define NLAYERS 3

typedef __attribute__((ext_vector_type(2))) float v2f;
typedef __attribute__((ext_vector_type(8))) float v8f;

// Compute one 16x16 f32 output tile of A[MxK=...x128] @ W[128x128] via
// V_WMMA_F32_16X16X4_F32, 32 K-steps.
// ISA layouts (CDNA5 ISA §7.12.2):
//   A 16x4: lanes 0-15 hold {K=k0,k0+1}, lanes 16-31 hold {K=k0+2,k0+3}, M = lane%16
//   B 4x16: lanes 0-15 hold rows K=k0..k0+1, lanes 16-31 rows K=k0+2..k0+3, N = lane%16
//   C/D 16x16: VGPR v: lanes 0-15 -> M=v, lanes 16-31 -> M=v+8, N = lane%16
__device__ __forceinline__ v8f wmma_gemm_tile(const float* __restrict__ A,
                                              const float* __restrict__ W,
                                              int m, int n, int half) {
  v8f acc = {0.f, 0.f, 0.f, 0.f, 0.f, 0.f, 0.f, 0.f};
#pragma unroll
  for (int k0 = 0; k0 < FT; k0 += 4) {
    const int kk = k0 + half * 2;
    v2f a = *(const v2f*)(A + (size_t)m * FT + kk);  // contiguous, 8B aligned
    v2f b;
    b.x = W[(size_t)kk * FT + n];
    b.y = W[(size_t)(kk + 1) * FT + n];
    acc = __builtin_amdgcn_wmma_f32_16x16x4_f32(false, a, false, b,
                                                (short)0, acc, false, false);
  }
  return acc;
}

// ---------------- degree / normalization ----------------
__global__ void __launch_bounds__(256) degree_kernel(const int* __restrict__ dst,
                                                     float* __restrict__ deg, int nE) {
  int i = blockIdx.x * blockDim.x + threadIdx.x;
  if (i < nE) atomicAdd(&deg[dst[i]], 1.0f);
}

__global__ void __launch_bounds__(256) rsqrt_kernel(float* __restrict__ dis, int n) {
  int i = blockIdx.x * blockDim.x + threadIdx.x;
  if (i < n) dis[i] = rsqrtf(dis[i] + 1.0f);
}

// ---------------- proj GEMM: h = x @ W + b ----------------
__global__ void __launch_bounds__(256) gemm_proj_kernel(const float* __restrict__ x,
                                                        const float* __restrict__ W,
                                                        const float* __restrict__ bias,
                                                        float* __restrict__ h) {
  const int lane = threadIdx.x & 31;
  const int wave = threadIdx.x >> 5;
  const int half = lane >> 4;
  const int l16 = lane & 15;
  const int baseM = blockIdx.x * 16;
  const int baseN = wave * 16;
  v8f acc = wmma_gemm_tile(x, W, baseM + l16, baseN + l16, half);
  const int n = baseN + l16;
  const float bn = bias[n];
#pragma unroll
  for (int v = 0; v < 8; ++v) {
    const int r = baseM + v + half * 8;
    h[(size_t)r * FT + n] = acc[v] + bn;
  }
}

// ---------------- conv GEMM + fused epilogue ----------------
// hw = h @ W ; agg = hw*dis^2 + conv_b + h  (self-loop + bias + residual pre-add)
__global__ void __launch_bounds__(256) gemm_conv_kernel(const float* __restrict__ h,
                                                        const float* __restrict__ W,
                                                        const float* __restrict__ bias,
                                                        const float* __restrict__ dis,
                                                        float* __restrict__ hw,
                                                        float* __restrict__ agg) {
  const int lane = threadIdx.x & 31;
  const int wave = threadIdx.x >> 5;
  const int half = lane >> 4;
  const int l16 = lane & 15;
  const int baseM = blockIdx.x * 16;
  const int baseN = wave * 16;
  v8f acc = wmma_gemm_tile(h, W, baseM + l16, baseN + l16, half);
  const int n = baseN + l16;
  const float bn = bias[n];
#pragma unroll
  for (int v = 0; v < 8; ++v) {
    const int r = baseM + v + half * 8;
    const float dv = dis[r];
    const float val = acc[v];
    hw[(size_t)r * FT + n] = val;
    agg[(size_t)r * FT + n] = val * (dv * dv) + bn + h[(size_t)r * FT + n];
  }
}

// ---------------- edge scatter: one wave per edge ----------------
__global__ void __launch_bounds__(256) scatter_kernel(const float* __restrict__ hw,
                                                      const int* __restrict__ src,
                                                      const int* __restrict__ dst,
                                                      const float* __restrict__ dis,
                                                      float* __restrict__ agg, int nE) {
  const int lane = threadIdx.x & 31;
  const long long gw = ((long long)blockIdx.x * blockDim.x + threadIdx.x) >> 5;
  if (gw >= nE) return;
  const int s = src[gw];
  const int d = dst[gw];
  const float norm = dis[s] * dis[d];
  const float4 v = ((const float4*)(hw + (size_t)s * FT))[lane];
  float* a = agg + (size_t)d * FT + lane * 4;
  atomicAdd(a + 0, v.x * norm);
  atomicAdd(a + 1, v.y * norm);
  atomicAdd(a + 2, v.z * norm);
  atomicAdd(a + 3, v.w * norm);
}

// ---------------- activation: h = leaky_relu(agg, 0.01) ----------------
__global__ void __launch_bounds__(256) act_kernel(const float* __restrict__ agg,
                                                  float* __restrict__ h, int n4) {
  int i = blockIdx.x * blockDim.x + threadIdx.x;
  if (i >= n4) return;
  float4 v = ((const float4*)agg)[i];
  v.x = v.x >= 0.f ? v.x : 0.01f * v.x;
  v.y = v.y >= 0.f ? v.y : 0.01f * v.y;
  v.z = v.z >= 0.f ? v.z : 0.01f * v.z;
  v.w = v.w >= 0.f ? v.w : 0.01f * v.w;
  ((float4*)h)[i] = v;
}

// ---------------- output: out = prelu(h@linW+linb) + prelu(x@resW+resb) ----------------
__global__ void __launch_bounds__(256) gemm_out_kernel(const float* __restrict__ h,
                                                       const float* __restrict__ x,
                                                       const float* __restrict__ linW,
                                                       const float* __restrict__ linB,
                                                       const float* __restrict__ resW,
                                                       const float* __restrict__ resB,
                                                       const float* __restrict__ preluA,
                                                       float* __restrict__ out) {
  const int lane = threadIdx.x & 31;
  const int wave = threadIdx.x >> 5;
  const int half = lane >> 4;
  const int l16 = lane & 15;
  const int baseM = blockIdx.x * 16;
  const int baseN = wave * 16;
  const float p = *preluA;
  v8f acc1 = wmma_gemm_tile(h, linW, baseM + l16, baseN + l16, half);
  v8f acc2 = wmma_gemm_tile(x, resW, baseM + l16, baseN + l16, half);
  const int n = baseN + l16;
  const float b1 = linB[n];
  const float b2 = resB[n];
#pragma unroll
  for (int v = 0; v < 8; ++v) {
    const int r = baseM + v + half * 8;
    float v1 = acc1[v] + b1;
    float v2 = acc2[v] + b2;
    v1 = v1 >= 0.f ? v1 : p * v1;
    v2 = v2 >= 0.f ? v2 : p * v2;
    out[(size_t)r * FT + n] = v1 + v2;
  }
}

extern "C" void kernel_launch(void* const* d_in, const int* in_sizes, int n_in,
                              void* d_out, int out_size, void* d_ws, size_t ws_size,
                              hipStream_t stream) {
  const float* x      = (const float*)d_in[0];
  const int*   ei     = (const int*)d_in[1];
  const float* projW  = (const float*)d_in[2];
  const float* projB  = (const float*)d_in[3];
  const float* convW  = (const float*)d_in[4];
  const float* convB  = (const float*)d_in[5];
  const float* linW   = (const float*)d_in[6];
  const float* linB   = (const float*)d_in[7];
  const float* resW   = (const float*)d_in[8];
  const float* resB   = (const float*)d_in[9];
  const float* preluA = (const float*)d_in[10];
  float* out = (float*)d_out;

  const int nNodes = in_sizes[0] / FT;   // 100000 (multiple of 16)
  const int nEdges = in_sizes[1] / 2;    // 1600000
  const int* src  = ei;                  // edge_index row 0
  const int* dstp = ei + nEdges;         // edge_index row 1

  // Workspace layout
  char* ws = (char*)d_ws;
  float* dis = (float*)ws;                               // nNodes floats
  float* h   = (float*)(ws + (1u << 20));                // nNodes*FT floats
  float* hw  = h + (size_t)nNodes * FT;
  float* agg = hw + (size_t)nNodes * FT;

  const int mTiles = nNodes / 16;            // 6250
  const int n4 = nNodes * FT / 4;
  const long long scatterWaves = (long long)nEdges;
  const int scatterBlocks = (int)((scatterWaves * 32 + 255) / 256);

  // Degree + D^{-1/2}
  hipMemsetAsync(dis, 0, (size_t)nNodes * sizeof(float), stream);
  degree_kernel<<<(nEdges + 255) / 256, 256, 0, stream>>>(dstp, dis, nEdges);
  rsqrt_kernel<<<(nNodes + 255) / 256, 256, 0, stream>>>(dis, nNodes);

  // Projection
  gemm_proj_kernel<<<mTiles, 256, 0, stream>>>(x, projW, projB, h);

  // GCN layers
  for (int l = 0; l < NLAYERS; ++l) {
    gemm_conv_kernel<<<mTiles, 256, 0, stream>>>(
        h, convW + (size_t)l * FT * FT, convB + (size_t)l * FT, dis, hw, agg);
    scatter_kernel<<<scatterBlocks, 256, 0, stream>>>(hw, src, dstp, dis, agg, nEdges);
    act_kernel<<<(n4 + 255) / 256, 256, 0, stream>>>(agg, h, n4);
  }

  // Output heads (both GEMMs fused, PReLU epilogue)
  gemm_out_kernel<<<mTiles, 256, 0, stream>>>(h, x, linW, linB, resW, resB, preluA, out);
}